// FixedNet_56040733278664
// MI455X (gfx1250) — compile-verified
//
#include <hip/hip_runtime.h>

typedef float v2f __attribute__((ext_vector_type(2)));
typedef float v8f __attribute__((ext_vector_type(8)));

#define FEAT  512
#define NCOLS 516   // 512 features + 4 layer columns

__device__ __forceinline__ void gatomicAdd(float* p, float v) {
  (void)__hip_atomic_fetch_add(p, v, __ATOMIC_RELAXED, __HIP_MEMORY_SCOPE_AGENT);
}

// ---------------------------------------------------------------------------
// Stage A1: per-node dot products  dn0[n] = x[n,:]·Wn0,  ds0[n] = x[n,:]·Ws0
// One wave (32 lanes) per node; lane owns 16 contiguous features.
// ---------------------------------------------------------------------------
__global__ void __launch_bounds__(256)
dots_kernel(const float* __restrict__ x, const float* __restrict__ wn,
            const float* __restrict__ ws, float* __restrict__ dn,
            float* __restrict__ ds, int n)
{
  int node = blockIdx.x * (blockDim.x >> 5) + (threadIdx.x >> 5);
  int lane = threadIdx.x & 31;
  if (node >= n) return;  // wave-uniform
  const float4* xr  = (const float4*)(x + (size_t)node * FEAT);
  const float4* wn4 = (const float4*)wn;
  const float4* ws4 = (const float4*)ws;
  float pn = 0.f, ps = 0.f;
#pragma unroll
  for (int j = 0; j < 4; ++j) {
    float4 v = xr[lane * 4 + j];
    float4 a = wn4[lane * 4 + j];
    float4 b = ws4[lane * 4 + j];
    pn += v.x * a.x + v.y * a.y + v.z * a.z + v.w * a.w;
    ps += v.x * b.x + v.y * b.y + v.z * b.z + v.w * b.w;
  }
#pragma unroll
  for (int off = 16; off > 0; off >>= 1) {
    pn += __shfl_xor(pn, off, 32);
    ps += __shfl_xor(ps, off, 32);
  }
  if (lane == 0) { dn[node] = pn; ds[node] = ps; }
}

// ---------------------------------------------------------------------------
// Stage A2: out[g, 0:512] += segment_sum(x) as a one-hot GEMM via
// V_WMMA_F32_16X16X4_F32.  A = 16 graphs x 4 nodes (one-hot membership),
// B = 4 nodes x 16 features of x, D/C = 16x16 f32 accumulator.
// One wave per (16-feature tile, node strip). graph_ids sorted -> the
// 16-graph group window per 4-node chunk is computed wave-uniformly.
// ---------------------------------------------------------------------------
__global__ void __launch_bounds__(32)
graphsum_x_wmma(const float* __restrict__ x, const int* __restrict__ gid,
                float* __restrict__ out, int n, int strip)
{
  const int fbase = blockIdx.x << 4;
  int s0 = blockIdx.y * strip;
  int s1 = s0 + strip; if (s1 > n) s1 = n;
  const int lane = threadIdx.x;
  const int l16  = lane & 15;
  const int half = lane >> 4;

  v8f acc0 = {}, acc1 = {}, acc2 = {}, acc3 = {};

  for (int c = s0; c + 3 < s1; c += 4) {
    // B: lanes 0-15 hold nodes c,c+1 (K=0,1); lanes 16-31 hold c+2,c+3 (K=2,3)
    const int na = c + (half << 1);
    const float* xp = x + (size_t)na * FEAT + fbase + l16;
    v2f B; B.x = xp[0]; B.y = xp[FEAT];

    const int ga = gid[na];
    const int gb = gid[na + 1];
    // wave-uniform group window (gid sorted => gid[c] <= chunk <= gid[c+3])
    const int gFirst = gid[c] >> 4;
    const int gLast  = gid[c + 3] >> 4;

    v2f A;
#define DO_GRP(G, ACC)                                                        \
    if (gFirst <= (G) && (G) <= gLast) {                                      \
      A.x = (ga == (((G) << 4) + l16)) ? 1.0f : 0.0f;                         \
      A.y = (gb == (((G) << 4) + l16)) ? 1.0f : 0.0f;                         \
      ACC = __builtin_amdgcn_wmma_f32_16x16x4_f32(                            \
          false, A, false, B, (short)0, ACC, false, false);                   \
    }
    DO_GRP(0, acc0)
    DO_GRP(1, acc1)
    DO_GRP(2, acc2)
    DO_GRP(3, acc3)
#undef DO_GRP
  }

  // Flush: D layout -> lane l16 = feature col, VGPR r = graph row r (+8 for half 1)
#define FLUSH(G, ACC)                                                         \
  _Pragma("unroll")                                                           \
  for (int r = 0; r < 8; ++r) {                                               \
    float v = ACC[r];                                                         \
    if (v != 0.0f) {                                                          \
      int graph = ((G) << 4) + (half << 3) + r;                               \
      gatomicAdd(out + (size_t)graph * NCOLS + fbase + l16, v);               \
    }                                                                         \
  }
  FLUSH(0, acc0)
  FLUSH(1, acc1)
  FLUSH(2, acc2)
  FLUSH(3, acc3)
#undef FLUSH
}

// ---------------------------------------------------------------------------
// Edge scatter: acc[dst[e]] += proj[src[e]]  (4 edges per thread, int4 loads)
// ---------------------------------------------------------------------------
__global__ void __launch_bounds__(256)
edge_scatter(const int* __restrict__ src, const int* __restrict__ dst,
             const float* __restrict__ proj, float* __restrict__ acc, int nE)
{
  int e = (blockIdx.x * blockDim.x + threadIdx.x) << 2;
  if (e + 3 < nE) {
    int4 s = *(const int4*)(src + e);
    int4 d = *(const int4*)(dst + e);
    gatomicAdd(acc + d.x, proj[s.x]);
    gatomicAdd(acc + d.y, proj[s.y]);
    gatomicAdd(acc + d.z, proj[s.z]);
    gatomicAdd(acc + d.w, proj[s.w]);
  } else {
    for (; e < nE; ++e) gatomicAdd(acc + dst[e], proj[src[e]]);
  }
}

// ---------------------------------------------------------------------------
// Node update: h = relu(acc + b + self*ws); also pre-scale next layer's proj
// ---------------------------------------------------------------------------
__global__ void __launch_bounds__(256)
node_update(const float* __restrict__ acc, const float* __restrict__ self_arr,
            const float* __restrict__ w_self, const float* __restrict__ b,
            float* __restrict__ h_out, float* __restrict__ proj_out,
            const float* __restrict__ w_neigh_next, int n)
{
  int i = blockIdx.x * blockDim.x + threadIdx.x;
  if (i >= n) return;
  float wsv = w_self ? w_self[0] : 1.0f;
  float h = acc[i] + b[0] + self_arr[i] * wsv;
  h = fmaxf(h, 0.0f);
  h_out[i] = h;
  if (proj_out) proj_out[i] = h * w_neigh_next[0];
}

// ---------------------------------------------------------------------------
// Per-graph sums of h1..h4 into out columns 512..515 via LDS ds_add_f32
// ---------------------------------------------------------------------------
__global__ void __launch_bounds__(256)
hsum_kernel(const float* __restrict__ h1, const float* __restrict__ h2,
            const float* __restrict__ h3, const float* __restrict__ h4,
            const int* __restrict__ gid, float* __restrict__ out, int n)
{
  __shared__ float lds[256];  // 64 graphs x 4 layer columns
  int t = threadIdx.x;
  lds[t] = 0.0f;
  __syncthreads();
  int i = blockIdx.x * blockDim.x + t;
  if (i < n) {
    int g = gid[i] << 2;
    (void)__hip_atomic_fetch_add(&lds[g + 0], h1[i], __ATOMIC_RELAXED, __HIP_MEMORY_SCOPE_WORKGROUP);
    (void)__hip_atomic_fetch_add(&lds[g + 1], h2[i], __ATOMIC_RELAXED, __HIP_MEMORY_SCOPE_WORKGROUP);
    (void)__hip_atomic_fetch_add(&lds[g + 2], h3[i], __ATOMIC_RELAXED, __HIP_MEMORY_SCOPE_WORKGROUP);
    (void)__hip_atomic_fetch_add(&lds[g + 3], h4[i], __ATOMIC_RELAXED, __HIP_MEMORY_SCOPE_WORKGROUP);
  }
  __syncthreads();
  float v = lds[t];
  if (v != 0.0f) {
    int graph = t >> 2;
    int col   = 512 + (t & 3);
    gatomicAdd(out + (size_t)graph * NCOLS + col, v);
  }
}

// ---------------------------------------------------------------------------
extern "C" void kernel_launch(void* const* d_in, const int* in_sizes, int n_in,
                              void* d_out, int out_size, void* d_ws, size_t ws_size,
                              hipStream_t stream)
{
  (void)n_in; (void)ws_size;
  const float* x    = (const float*)d_in[0];
  const int*   esrc = (const int*)d_in[1];
  const int*   edst = (const int*)d_in[2];
  const int*   gid  = (const int*)d_in[3];
  const float* Wn0  = (const float*)d_in[4];
  const float* Ws0  = (const float*)d_in[5];
  const float* b0   = (const float*)d_in[6];
  const float* WnR  = (const float*)d_in[7];  // [3]
  const float* WsR  = (const float*)d_in[8];  // [3]
  const float* bR   = (const float*)d_in[9];  // [3]
  float* out = (float*)d_out;

  const int N = in_sizes[0] / FEAT;
  const int E = in_sizes[1];

  float* ws  = (float*)d_ws;
  float* dn0  = ws + 0 * (size_t)N;
  float* ds0  = ws + 1 * (size_t)N;
  float* acc  = ws + 2 * (size_t)N;
  float* proj = ws + 3 * (size_t)N;
  float* h1   = ws + 4 * (size_t)N;
  float* h2   = ws + 5 * (size_t)N;
  float* h3   = ws + 6 * (size_t)N;
  float* h4   = ws + 7 * (size_t)N;

  hipMemsetAsync(out, 0, (size_t)out_size * sizeof(float), stream);

  // Stage A1: projections of x
  dots_kernel<<<(N + 7) / 8, 256, 0, stream>>>(x, Wn0, Ws0, dn0, ds0, N);

  // Stage A2: per-graph feature sums via WMMA one-hot GEMM
  const int STRIP = 1024;  // multiple of 4
  dim3 g2(FEAT / 16, (N + STRIP - 1) / STRIP);
  graphsum_x_wmma<<<g2, 32, 0, stream>>>(x, gid, out, N, STRIP);

  const int eBlocks = (((E + 3) / 4) + 255) / 256;
  const int nBlocks = (N + 255) / 256;

  // Layer 0: proj = dn0, self = ds0 (unscaled), bias b0 -> h1, proj = h1*WnR[0]
  hipMemsetAsync(acc, 0, (size_t)N * sizeof(float), stream);
  edge_scatter<<<eBlocks, 256, 0, stream>>>(esrc, edst, dn0, acc, E);
  node_update<<<nBlocks, 256, 0, stream>>>(acc, ds0, nullptr, b0, h1, proj, WnR + 0, N);

  // Layer 1 -> h2, proj = h2*WnR[1]
  hipMemsetAsync(acc, 0, (size_t)N * sizeof(float), stream);
  edge_scatter<<<eBlocks, 256, 0, stream>>>(esrc, edst, proj, acc, E);
  node_update<<<nBlocks, 256, 0, stream>>>(acc, h1, WsR + 0, bR + 0, h2, proj, WnR + 1, N);

  // Layer 2 -> h3, proj = h3*WnR[2]
  hipMemsetAsync(acc, 0, (size_t)N * sizeof(float), stream);
  edge_scatter<<<eBlocks, 256, 0, stream>>>(esrc, edst, proj, acc, E);
  node_update<<<nBlocks, 256, 0, stream>>>(acc, h2, WsR + 1, bR + 1, h3, proj, WnR + 2, N);

  // Layer 3 -> h4 (no next proj)
  hipMemsetAsync(acc, 0, (size_t)N * sizeof(float), stream);
  edge_scatter<<<eBlocks, 256, 0, stream>>>(esrc, edst, proj, acc, E);
  node_update<<<nBlocks, 256, 0, stream>>>(acc, h3, WsR + 2, bR + 2, h4, nullptr, nullptr, N);

  // Final: per-graph sums of h1..h4 -> out[:, 512:516]
  hsum_kernel<<<nBlocks, 256, 0, stream>>>(h1, h2, h3, h4, gid, out, N);
}